// TransMIL_6030134083788
// MI455X (gfx1250) — compile-verified
//
#include <hip/hip_runtime.h>
#include <hip/hip_bf16.h>
#include <math.h>

typedef __attribute__((ext_vector_type(2))) float v2f;
typedef __attribute__((ext_vector_type(8))) float v8f;

// ---------------- constants (from reference) ----------------
#define DIMC   512
#define HEADSC 8
#define DHC    64
#define MLAND  17
#define NPADT  2125   // Np = tokens padded to multiple of 17
#define NTOK   2117   // 1 cls + 2116
#define PADF   8      // front pad
#define LSEG   125    // Np / 17
#define HSQ    46     // ceil(sqrt(2048))
#define BATCH  2
#define BHC    16     // BATCH*HEADS

// ================= WMMA f32 GEMM =================
// C[M,N] = A[M,K] * op(B) (+bias) (+relu), row-major, batched via blockIdx.z.
// Block: 256 thr = 8 waves, tile 64(M) x 128(N), wave = 32x32 (2x2 wmma tiles).
#define BM 64
#define BN 128
#define BK 16

__device__ __forceinline__ void store_tile(v8f acc, float* __restrict__ C,
                                           int ldc, int M, int N, int gm0, int gn,
                                           const float* __restrict__ bias,
                                           int act, int nt, int half) {
  if (gn >= N) return;
  float bv = bias ? bias[gn] : 0.0f;
#pragma unroll
  for (int r = 0; r < 8; ++r) {
    int gm = gm0 + r + 8 * half;
    if (gm < M) {
      float val = acc[r] + bv;
      if (act) val = fmaxf(val, 0.0f);
      float* p = C + (long long)gm * ldc + gn;
      if (nt) __builtin_nontemporal_store(val, p);
      else    *p = val;
    }
  }
}

__global__ __launch_bounds__(256)
void gemm_f32_wmma(const float* __restrict__ A, const float* __restrict__ B,
                   float* __restrict__ C, const float* __restrict__ bias,
                   int M, int N, int K, int lda, int ldb, int ldc,
                   long long sA, long long sB, long long sC,
                   int transB, int act, int nt) {
  __shared__ float As[BM][BK];       // 4 KB
  __shared__ float Bs[BK][BN + 4];   // 8.25 KB (pad vs bank conflicts)
  int bz = blockIdx.z;
  A += sA * bz; B += sB * bz; C += sC * bz;
  int m0 = blockIdx.y * BM;
  int n0 = blockIdx.x * BN;
  int t = threadIdx.x;
  int lane = t & 31, wid = t >> 5;
  int wm = (wid >> 2) * 32;  // 0 / 32
  int wn = (wid & 3) * 32;   // 0..96
  int half = lane >> 4;      // which 16-lane half (K pair select)
  int lr = lane & 15;        // row (A) / col (B,C)

  // block-uniform fast-path eligibility (alignment + interior tile)
  const bool aAligned = ((lda & 3) == 0);
  const bool bAligned = (!transB) && ((ldb & 3) == 0);
  const bool mFull = (m0 + BM <= M);
  const bool nFull = (n0 + BN <= N);

  v8f acc00 = {}; v8f acc01 = {}; v8f acc10 = {}; v8f acc11 = {};

  for (int k0 = 0; k0 < K; k0 += BK) {
    const bool kFull = (k0 + BK <= K);
    // ---- stage A tile: 64x16 ----
    if (aAligned && mFull && kFull) {
      // one float4 per thread: row = t>>2, k4 = (t&3)*4  -> global_load_b128
      int row = t >> 2, k4 = (t & 3) * 4;
      const float* ap = A + (long long)(m0 + row) * lda + (k0 + k4);
      float4 av = *(const float4*)ap;
      As[row][k4]     = av.x; As[row][k4 + 1] = av.y;
      As[row][k4 + 2] = av.z; As[row][k4 + 3] = av.w;
      if (k0 + BK < K) __builtin_prefetch(ap + BK, 0, 1);
    } else {
#pragma unroll
      for (int i = 0; i < 4; ++i) {          // 1024 elems, 4/thread, guarded
        int idx = t + 256 * i;
        int r = idx >> 4, kk = idx & 15;
        int gm = m0 + r, gk = k0 + kk;
        As[r][kk] = (gm < M && gk < K) ? A[(long long)gm * lda + gk] : 0.0f;
      }
    }
    // ---- stage B tile: 16x128 ----
    if (bAligned && nFull && kFull) {
      // two float4 per thread -> global_load_b128 + ds_store_b128
#pragma unroll
      for (int i = 0; i < 2; ++i) {
        int f4 = t + 256 * i;
        int kk = f4 >> 5, c4 = (f4 & 31) * 4;
        const float* bp = B + (long long)(k0 + kk) * ldb + (n0 + c4);
        float4 bv = *(const float4*)bp;
        Bs[kk][c4]     = bv.x; Bs[kk][c4 + 1] = bv.y;
        Bs[kk][c4 + 2] = bv.z; Bs[kk][c4 + 3] = bv.w;
        if (k0 + BK < K) __builtin_prefetch(bp + (long long)BK * ldb, 0, 1);
      }
    } else {
#pragma unroll
      for (int i = 0; i < 8; ++i) {          // 2048 elems, 8/thread, guarded
        int idx = t + 256 * i;
        int kk = idx >> 7, c = idx & 127;
        int gk = k0 + kk, gn = n0 + c;
        float val = 0.0f;
        if (gk < K && gn < N)
          val = transB ? B[(long long)gn * ldb + gk] : B[(long long)gk * ldb + gn];
        Bs[kk][c] = val;
      }
    }
    __syncthreads();
#pragma unroll
    for (int kk = 0; kk < BK; kk += 4) {
      int ka = kk + 2 * half;
      v2f a0, a1v, b0, b1v;
      // A 16x4 layout: lanes 0-15 rows M, VGPR0={K0|K2}, VGPR1={K1|K3}
      a0.x  = As[wm + lr][ka];      a0.y  = As[wm + lr][ka + 1];
      a1v.x = As[wm + 16 + lr][ka]; a1v.y = As[wm + 16 + lr][ka + 1];
      // B 4x16 layout: lanes = N cols, VGPR0={K0|K2}, VGPR1={K1|K3}
      b0.x  = Bs[ka][wn + lr];      b0.y  = Bs[ka + 1][wn + lr];
      b1v.x = Bs[ka][wn + 16 + lr]; b1v.y = Bs[ka + 1][wn + 16 + lr];
      acc00 = __builtin_amdgcn_wmma_f32_16x16x4_f32(false, a0,  false, b0,  (short)0, acc00, false, false);
      acc01 = __builtin_amdgcn_wmma_f32_16x16x4_f32(false, a0,  false, b1v, (short)0, acc01, false, false);
      acc10 = __builtin_amdgcn_wmma_f32_16x16x4_f32(false, a1v, false, b0,  (short)0, acc10, false, false);
      acc11 = __builtin_amdgcn_wmma_f32_16x16x4_f32(false, a1v, false, b1v, (short)0, acc11, false, false);
    }
    __syncthreads();
  }
  // C/D 16x16 layout: VGPR r = rows {r | r+8}, lanes = cols
  store_tile(acc00, C, ldc, M, N, m0 + wm,      n0 + wn + lr,      bias, act, nt, half);
  store_tile(acc01, C, ldc, M, N, m0 + wm,      n0 + wn + 16 + lr, bias, act, nt, half);
  store_tile(acc10, C, ldc, M, N, m0 + wm + 16, n0 + wn + lr,      bias, act, nt, half);
  store_tile(acc11, C, ldc, M, N, m0 + wm + 16, n0 + wn + 16 + lr, bias, act, nt, half);
}

// ================= token assembly =================
// T[b,0]=cls ; T[b,1+i]=h[b, i mod 2048]  (wrap pad to 46x46)
__global__ void build_tokens(const float* __restrict__ h, const float* __restrict__ cls,
                             float* __restrict__ T) {
  int j = blockIdx.x, b = blockIdx.y;
  const float* src;
  if (j == 0) src = cls;
  else {
    int i = j - 1; if (i >= 2048) i -= 2048;
    src = h + ((long long)b * 2048 + i) * DIMC;
  }
  float* dst = T + ((long long)b * NTOK + j) * DIMC;
  for (int c = threadIdx.x; c < DIMC; c += blockDim.x) dst[c] = src[c];
}

// ================= layernorm + front-pad into xp (Np rows) =================
__global__ void ln_pad(const float* __restrict__ T, const float* __restrict__ g,
                       const float* __restrict__ b, float* __restrict__ xp) {
  __shared__ float red[256];
  int j = blockIdx.x, bb = blockIdx.y, t = threadIdx.x;
  float* dst = xp + ((long long)bb * NPADT + j) * DIMC;
  if (j < PADF) { for (int c = t; c < DIMC; c += 256) dst[c] = 0.0f; return; }
  const float* x = T + ((long long)bb * NTOK + (j - PADF)) * DIMC;
  float s = 0.0f;
  for (int c = t; c < DIMC; c += 256) s += x[c];
  red[t] = s; __syncthreads();
  for (int k = 128; k > 0; k >>= 1) { if (t < k) red[t] += red[t + k]; __syncthreads(); }
  float mu = red[0] * (1.0f / DIMC); __syncthreads();
  float v = 0.0f;
  for (int c = t; c < DIMC; c += 256) { float d = x[c] - mu; v += d * d; }
  red[t] = v; __syncthreads();
  for (int k = 128; k > 0; k >>= 1) { if (t < k) red[t] += red[t + k]; __syncthreads(); }
  float rs = rsqrtf(red[0] * (1.0f / DIMC) + 1e-5f); __syncthreads();
  for (int c = t; c < DIMC; c += 256) dst[c] = (x[c] - mu) * rs * g[c] + b[c];
}

// ================= split qkv -> head-major q/k/v (q scaled by DH^-0.5) =====
__global__ void split_heads(const float* __restrict__ qkv, float* __restrict__ q,
                            float* __restrict__ k, float* __restrict__ v) {
  int i = blockIdx.x, bb = blockIdx.y, c = threadIdx.x;  // c in [0,512)
  int h = c >> 6, d = c & 63;
  long long src = ((long long)bb * NPADT + i) * (3 * DIMC);
  long long dst = (((long long)bb * HEADSC + h) * NPADT + i) * DHC + d;
  q[dst] = qkv[src + c] * 0.125f;            // DH^-0.5 = 1/8
  k[dst] = qkv[src + DIMC + c];
  v[dst] = qkv[src + 2 * DIMC + c];
}

// ================= landmarks: mean over l=125 contiguous rows =============
__global__ void landmarks(const float* __restrict__ q, float* __restrict__ ql) {
  int m = blockIdx.x, bh = blockIdx.y, d = threadIdx.x;
  const float* src = q + ((long long)bh * NPADT + (long long)m * LSEG) * DHC + d;
  float s = 0.0f;
  for (int j = 0; j < LSEG; ++j) s += src[(long long)j * DHC];
  ql[((long long)bh * MLAND + m) * DHC + d] = s * (1.0f / LSEG);
}

// ================= row softmax ============================================
__global__ void softmax_rows(float* __restrict__ x, int L) {
  __shared__ float red[256];
  long long r = blockIdx.x;
  float* p = x + r * (long long)L;
  int t = threadIdx.x;
  float m = -3.402823466e38f;
  for (int c = t; c < L; c += 256) m = fmaxf(m, p[c]);
  red[t] = m; __syncthreads();
  for (int s = 128; s > 0; s >>= 1) { if (t < s) red[t] = fmaxf(red[t], red[t + s]); __syncthreads(); }
  m = red[0]; __syncthreads();
  float sum = 0.0f;
  for (int c = t; c < L; c += 256) sum += expf(p[c] - m);
  red[t] = sum; __syncthreads();
  for (int s = 128; s > 0; s >>= 1) { if (t < s) red[t] += red[t + s]; __syncthreads(); }
  float inv = 1.0f / red[0]; __syncthreads();
  for (int c = t; c < L; c += 256) p[c] = expf(p[c] - m) * inv;
}

// ================= pinv: global max(|rowsum|), max(|colsum|) ==============
__global__ void pinv_norm(const float* __restrict__ a2, float* __restrict__ nrm) {
  __shared__ float red[256];
  int t = threadIdx.x;
  float rm = 0.0f;
  for (int idx = t; idx < BHC * MLAND; idx += 256) {
    int bh = idx / MLAND, i = idx % MLAND;
    const float* p = a2 + bh * 289 + i * MLAND;
    float s = 0.0f;
    for (int j = 0; j < MLAND; ++j) s += fabsf(p[j]);
    rm = fmaxf(rm, s);
  }
  red[t] = rm; __syncthreads();
  for (int s = 128; s > 0; s >>= 1) { if (t < s) red[t] = fmaxf(red[t], red[t + s]); __syncthreads(); }
  if (t == 0) nrm[0] = red[0];
  __syncthreads();
  float cm = 0.0f;
  for (int idx = t; idx < BHC * MLAND; idx += 256) {
    int bh = idx / MLAND, j = idx % MLAND;
    const float* p = a2 + bh * 289 + j;
    float s = 0.0f;
    for (int i = 0; i < MLAND; ++i) s += fabsf(p[i * MLAND]);
    cm = fmaxf(cm, s);
  }
  red[t] = cm; __syncthreads();
  for (int s = 128; s > 0; s >>= 1) { if (t < s) red[t] = fmaxf(red[t], red[t + s]); __syncthreads(); }
  if (t == 0) nrm[1] = red[0];
}

// ================= pinv iterations (per (b,h), 17x17 threads) =============
__global__ void pinv_iter(const float* __restrict__ a2, const float* __restrict__ nrm,
                          float* __restrict__ a2i) {
  __shared__ float x[MLAND][MLAND + 1], z[MLAND][MLAND + 1];
  __shared__ float xz[MLAND][MLAND + 1], tm[MLAND][MLAND + 1];
  int i = threadIdx.y, j = threadIdx.x, bh = blockIdx.x;
  float inv = 1.0f / (nrm[0] * nrm[1]);
  x[i][j] = a2[(bh * MLAND + i) * MLAND + j];
  z[i][j] = a2[(bh * MLAND + j) * MLAND + i] * inv;   // x^T * inv
  __syncthreads();
  for (int it = 0; it < 6; ++it) {
    float s = 0.0f;
    for (int kk = 0; kk < MLAND; ++kk) s += x[i][kk] * z[kk][j];
    xz[i][j] = s; __syncthreads();
    tm[i][j] = (i == j ? 7.0f : 0.0f) - xz[i][j]; __syncthreads();
    s = 0.0f;
    for (int kk = 0; kk < MLAND; ++kk) s += xz[i][kk] * tm[kk][j];
    __syncthreads();
    tm[i][j] = (i == j ? 15.0f : 0.0f) - s; __syncthreads();
    s = 0.0f;
    for (int kk = 0; kk < MLAND; ++kk) s += xz[i][kk] * tm[kk][j];
    __syncthreads();
    tm[i][j] = (i == j ? 13.0f : 0.0f) - s; __syncthreads();
    s = 0.0f;
    for (int kk = 0; kk < MLAND; ++kk) s += z[i][kk] * tm[kk][j];
    __syncthreads();
    z[i][j] = 0.25f * s; __syncthreads();
  }
  a2i[(bh * MLAND + i) * MLAND + j] = z[i][j];
}

// ================= depthwise conv1d (k=33) over sequence, add into outh ===
__global__ void res_conv_add(const float* __restrict__ v, const float* __restrict__ w,
                             float* __restrict__ outh) {
  int i = blockIdx.x, bh = blockIdx.y, d = threadIdx.x, h = bh & 7;
  const float* vb = v + (long long)bh * NPADT * DHC;
  float s = 0.0f;
  for (int tt = 0; tt < 33; ++tt) {
    int src = i + tt - 16;
    if (src >= 0 && src < NPADT) s += vb[(long long)src * DHC + d] * w[h * 33 + tt];
  }
  outh[((long long)bh * NPADT + i) * DHC + d] += s;
}

// ================= merge heads back to (b, Np, 512) =======================
__global__ void merge_heads(const float* __restrict__ outh, float* __restrict__ mrg) {
  int i = blockIdx.x, bb = blockIdx.y, c = threadIdx.x;
  int h = c >> 6, d = c & 63;
  mrg[((long long)bb * NPADT + i) * DIMC + c] =
      outh[(((long long)bb * HEADSC + h) * NPADT + i) * DHC + d];
}

// ================= residual add: T += proj[:, -NTOK:] =====================
__global__ void resid_add(float* __restrict__ T, const float* __restrict__ proj) {
  int j = blockIdx.x, bb = blockIdx.y;
  float* dst = T + ((long long)bb * NTOK + j) * DIMC;
  const float* src = proj + ((long long)bb * NPADT + PADF + j) * DIMC;
  for (int c = threadIdx.x; c < DIMC; c += blockDim.x) dst[c] += src[c];
}

// ================= PPEG: fused dw7 + id + dw5 + dw3 on 46x46 ==============
__global__ void ppeg_kernel(const float* __restrict__ T, float* __restrict__ T2,
                            const float* __restrict__ w7, const float* __restrict__ b7,
                            const float* __restrict__ w5, const float* __restrict__ b5,
                            const float* __restrict__ w3, const float* __restrict__ b3) {
  int p = blockIdx.x, bb = blockIdx.y;
  int y = p / HSQ, x = p % HSQ;
  const float* base = T + ((long long)bb * NTOK + 1) * DIMC;
  for (int c = threadIdx.x; c < DIMC; c += blockDim.x) {
    float acc = base[(long long)p * DIMC + c] + b7[c] + b5[c] + b3[c];
    for (int dy = -3; dy <= 3; ++dy) {
      int yy = y + dy; if (yy < 0 || yy >= HSQ) continue;
      for (int dx = -3; dx <= 3; ++dx) {
        int xx = x + dx; if (xx < 0 || xx >= HSQ) continue;
        float pv = base[((long long)yy * HSQ + xx) * DIMC + c];
        acc += pv * w7[c * 49 + (dy + 3) * 7 + (dx + 3)];
        if (dy >= -2 && dy <= 2 && dx >= -2 && dx <= 2)
          acc += pv * w5[c * 25 + (dy + 2) * 5 + (dx + 2)];
        if (dy >= -1 && dy <= 1 && dx >= -1 && dx <= 1)
          acc += pv * w3[c * 9 + (dy + 1) * 3 + (dx + 1)];
      }
    }
    T2[((long long)bb * NTOK + 1 + p) * DIMC + c] = acc;
  }
}

__global__ void copy_cls(const float* __restrict__ T, float* __restrict__ T2) {
  int bb = blockIdx.x;
  for (int c = threadIdx.x; c < DIMC; c += blockDim.x)
    T2[(long long)bb * NTOK * DIMC + c] = T[(long long)bb * NTOK * DIMC + c];
}

// ================= final LN(row 0) + fc2 ==================================
__global__ void final_head(const float* __restrict__ T, const float* __restrict__ g,
                           const float* __restrict__ b, const float* __restrict__ w,
                           const float* __restrict__ fb, float* __restrict__ out) {
  __shared__ float red[256];
  int bb = blockIdx.x, t = threadIdx.x;
  const float* x = T + (long long)bb * NTOK * DIMC;
  float s = 0.0f;
  for (int c = t; c < DIMC; c += 256) s += x[c];
  red[t] = s; __syncthreads();
  for (int k = 128; k > 0; k >>= 1) { if (t < k) red[t] += red[t + k]; __syncthreads(); }
  float mu = red[0] * (1.0f / DIMC); __syncthreads();
  float v = 0.0f;
  for (int c = t; c < DIMC; c += 256) { float d = x[c] - mu; v += d * d; }
  red[t] = v; __syncthreads();
  for (int k = 128; k > 0; k >>= 1) { if (t < k) red[t] += red[t + k]; __syncthreads(); }
  float rs = rsqrtf(red[0] * (1.0f / DIMC) + 1e-5f); __syncthreads();
  float p0 = 0.0f, p1 = 0.0f;
  for (int c = t; c < DIMC; c += 256) {
    float xn = (x[c] - mu) * rs * g[c] + b[c];
    p0 += xn * w[c * 2 + 0]; p1 += xn * w[c * 2 + 1];
  }
  red[t] = p0; __syncthreads();
  for (int k = 128; k > 0; k >>= 1) { if (t < k) red[t] += red[t + k]; __syncthreads(); }
  if (t == 0) out[bb * 2 + 0] = red[0] + fb[0];
  __syncthreads();
  red[t] = p1; __syncthreads();
  for (int k = 128; k > 0; k >>= 1) { if (t < k) red[t] += red[t + k]; __syncthreads(); }
  if (t == 0) out[bb * 2 + 1] = red[0] + fb[1];
}

// =========================================================================
extern "C" void kernel_launch(void* const* d_in, const int* in_sizes, int n_in,
                              void* d_out, int out_size, void* d_ws, size_t ws_size,
                              hipStream_t stream) {
  (void)in_sizes; (void)n_in; (void)out_size; (void)ws_size;
  const float* feats   = (const float*)d_in[0];
  const float* fc1_w   = (const float*)d_in[1];
  const float* fc1_b   = (const float*)d_in[2];
  const float* cls_tok = (const float*)d_in[3];
  const float* l_ng[2]   = {(const float*)d_in[4],  (const float*)d_in[10]};
  const float* l_nb[2]   = {(const float*)d_in[5],  (const float*)d_in[11]};
  const float* l_qkvw[2] = {(const float*)d_in[6],  (const float*)d_in[12]};
  const float* l_outw[2] = {(const float*)d_in[7],  (const float*)d_in[13]};
  const float* l_outb[2] = {(const float*)d_in[8],  (const float*)d_in[14]};
  const float* l_resw[2] = {(const float*)d_in[9],  (const float*)d_in[15]};
  const float* w7 = (const float*)d_in[16]; const float* b7 = (const float*)d_in[17];
  const float* w5 = (const float*)d_in[18]; const float* b5 = (const float*)d_in[19];
  const float* w3 = (const float*)d_in[20]; const float* b3 = (const float*)d_in[21];
  const float* norm_g = (const float*)d_in[22];
  const float* norm_b = (const float*)d_in[23];
  const float* fc2_w  = (const float*)d_in[24];
  const float* fc2_b  = (const float*)d_in[25];

  float* out = (float*)d_out;
  float* aw1 = out + 4;
  float* aw2 = out + 4 + (size_t)BHC * NPADT * NPADT;

  // ---- workspace carve-out ----
  float* W = (float*)d_ws;
  size_t off = 0;
  auto take = [&](size_t n) { float* p = W + off; off += (n + 63) & ~(size_t)63; return p; };
  float* h    = take((size_t)BATCH * 2048 * DIMC);
  float* T    = take((size_t)BATCH * NTOK * DIMC);
  float* T2   = take((size_t)BATCH * NTOK * DIMC);
  float* xp   = take((size_t)BATCH * NPADT * DIMC);
  float* qkv  = take((size_t)BATCH * NPADT * 3 * DIMC);
  float* q    = take((size_t)BHC * NPADT * DHC);
  float* k    = take((size_t)BHC * NPADT * DHC);
  float* v    = take((size_t)BHC * NPADT * DHC);
  float* ql   = take((size_t)BHC * MLAND * DHC);
  float* kl   = take((size_t)BHC * MLAND * DHC);
  float* a1   = take((size_t)BHC * NPADT * MLAND);
  float* a3   = take((size_t)BHC * MLAND * NPADT);
  float* a2   = take((size_t)BHC * 289);
  float* a2i  = take((size_t)BHC * 289);
  float* nrm  = take(64);
  float* lft  = take((size_t)BHC * NPADT * MLAND);
  float* a3v  = take((size_t)BHC * MLAND * DHC);
  float* outh = take((size_t)BHC * NPADT * DHC);
  float* mrg  = take((size_t)BATCH * NPADT * DIMC);
  float* proj = take((size_t)BATCH * NPADT * DIMC);

  auto gemm = [&](const float* A, const float* Bm, float* C, const float* bias,
                  int M, int N, int K, int lda, int ldb, int ldc,
                  long long sA, long long sB, long long sC,
                  int transB, int act, int nt, int batch) {
    dim3 g((unsigned)((N + BN - 1) / BN), (unsigned)((M + BM - 1) / BM), (unsigned)batch);
    gemm_f32_wmma<<<g, 256, 0, stream>>>(A, Bm, C, bias, M, N, K, lda, ldb, ldc,
                                         sA, sB, sC, transB, act, nt);
  };

  // 1) fc1 + ReLU : (B*2048, 1024) x (1024, 512)
  gemm(feats, fc1_w, h, fc1_b, BATCH * 2048, DIMC, 1024, 1024, DIMC, DIMC,
       0, 0, 0, 0, /*relu*/1, 0, 1);
  // 2) assemble tokens (wrap-pad + cls)
  build_tokens<<<dim3(NTOK, BATCH), 256, 0, stream>>>(h, cls_tok, T);

  auto run_layer = [&](int li, float* Tc, float* aw) {
    ln_pad<<<dim3(NPADT, BATCH), 256, 0, stream>>>(Tc, l_ng[li], l_nb[li], xp);
    gemm(xp, l_qkvw[li], qkv, nullptr, BATCH * NPADT, 3 * DIMC, DIMC,
         DIMC, 3 * DIMC, 3 * DIMC, 0, 0, 0, 0, 0, 0, 1);
    split_heads<<<dim3(NPADT, BATCH), 512, 0, stream>>>(qkv, q, k, v);
    landmarks<<<dim3(MLAND, BHC), DHC, 0, stream>>>(q, ql);
    landmarks<<<dim3(MLAND, BHC), DHC, 0, stream>>>(k, kl);
    // a1 = q @ kl^T   (Np x 17)
    gemm(q, kl, a1, nullptr, NPADT, MLAND, DHC, DHC, DHC, MLAND,
         (long long)NPADT * DHC, (long long)MLAND * DHC, (long long)NPADT * MLAND, 1, 0, 0, BHC);
    // a2 = ql @ kl^T  (17 x 17)
    gemm(ql, kl, a2, nullptr, MLAND, MLAND, DHC, DHC, DHC, MLAND,
         (long long)MLAND * DHC, (long long)MLAND * DHC, 289, 1, 0, 0, BHC);
    // a3 = ql @ k^T   (17 x Np)
    gemm(ql, k, a3, nullptr, MLAND, NPADT, DHC, DHC, DHC, NPADT,
         (long long)MLAND * DHC, (long long)NPADT * DHC, (long long)MLAND * NPADT, 1, 0, 0, BHC);
    softmax_rows<<<BHC * NPADT, 256, 0, stream>>>(a1, MLAND);
    softmax_rows<<<BHC * MLAND, 256, 0, stream>>>(a2, MLAND);
    softmax_rows<<<BHC * MLAND, 256, 0, stream>>>(a3, NPADT);
    pinv_norm<<<1, 256, 0, stream>>>(a2, nrm);
    pinv_iter<<<BHC, dim3(MLAND, MLAND), 0, stream>>>(a2, nrm, a2i);
    // left = a1 @ a2i (Np x 17)
    gemm(a1, a2i, lft, nullptr, NPADT, MLAND, MLAND, MLAND, MLAND, MLAND,
         (long long)NPADT * MLAND, 289, (long long)NPADT * MLAND, 0, 0, 0, BHC);
    // a3v = a3 @ v    (17 x 64)
    gemm(a3, v, a3v, nullptr, MLAND, DHC, NPADT, NPADT, DHC, DHC,
         (long long)MLAND * NPADT, (long long)NPADT * DHC, (long long)MLAND * DHC, 0, 0, 0, BHC);
    // outh = left @ a3v (Np x 64)
    gemm(lft, a3v, outh, nullptr, NPADT, DHC, MLAND, MLAND, DHC, DHC,
         (long long)NPADT * MLAND, (long long)MLAND * DHC, (long long)NPADT * DHC, 0, 0, 0, BHC);
    res_conv_add<<<dim3(NPADT, BHC), DHC, 0, stream>>>(v, l_resw[li], outh);
    // attn map = left @ a3 (Np x Np) -> streamed to d_out, NT stores
    gemm(lft, a3, aw, nullptr, NPADT, NPADT, MLAND, MLAND, NPADT, NPADT,
         (long long)NPADT * MLAND, (long long)MLAND * NPADT, (long long)NPADT * NPADT,
         0, 0, /*nt*/1, BHC);
    merge_heads<<<dim3(NPADT, BATCH), 512, 0, stream>>>(outh, mrg);
    gemm(mrg, l_outw[li], proj, l_outb[li], BATCH * NPADT, DIMC, DIMC,
         DIMC, DIMC, DIMC, 0, 0, 0, 0, 0, 0, 1);
    resid_add<<<dim3(NTOK, BATCH), 256, 0, stream>>>(Tc, proj);
  };

  // 3) layer 1
  run_layer(0, T, aw1);
  // 4) PPEG
  ppeg_kernel<<<dim3(HSQ * HSQ, BATCH), 256, 0, stream>>>(T, T2, w7, b7, w5, b5, w3, b3);
  copy_cls<<<BATCH, 256, 0, stream>>>(T, T2);
  // 5) layer 2
  run_layer(1, T2, aw2);
  // 6) final LN + fc2
  final_head<<<BATCH, 256, 0, stream>>>(T2, norm_g, norm_b, fc2_w, fc2_b, out);
}